// IdentTreeModeler_52295521796446
// MI455X (gfx1250) — compile-verified
//
#include <hip/hip_runtime.h>
#include <math.h>

typedef __attribute__((ext_vector_type(2))) float v2f;
typedef __attribute__((ext_vector_type(8))) float v8f;

#define E_DIM 1024
#define H_DIM 2048
#define R_DIM 1024
#define G_DIM 8192
#define RULES_N 50000

// workspace layout (float offsets); total ~259224 floats (~1.01 MB)
#define WS_X      0        // 1024   : x = emb + relu(rem @ W_rin + b)
#define WS_GATES  1024     // 8192   : LSTM pre-activations
#define WS_Z      9216     // 50000  : relu(h @ W_out + b)
#define WS_RED    59216    // 8      : [max, log(sum exp)]
#define WS_P      59224    // 200000 : 4 k-split partials of W_out matvec

// ---------------------------------------------------------------------------
// GEMV with W stored [K, N] row-major: out[n] = addend[n] + relu(sum_k v[k]*W[k,n] + bias[n])
// Uses V_WMMA_F32_16X16X4_F32: A(16x4) = v broadcast over M rows, B(4x16) = W tile.
// Every row of D is the same dot-product vector -> lane n<16, acc[0] = out col e0+n.
// B-tile loads are coalesced 64B segments per 16-lane half.
// ---------------------------------------------------------------------------
__global__ void gemv_kn_relu_kernel(const float* __restrict__ W,
                                    const float* __restrict__ vec,
                                    const float* __restrict__ bias,
                                    const float* __restrict__ emb,   // may be null
                                    const int*   __restrict__ ident, // may be null
                                    float* __restrict__ out, int K, int N)
{
    const int lane = threadIdx.x & 31;
    const int wave = threadIdx.x >> 5;
    const int tile = blockIdx.x * (blockDim.x >> 5) + wave;
    const int e0   = tile * 16;
    if (e0 >= N) return;              // wave-uniform: EXEC stays all-ones for WMMA
    const int half = lane >> 4;       // 0: K={k0,k0+1}, 1: K={k0+2,k0+3}
    const int ln   = lane & 15;
    const int col  = e0 + ln;

#if __has_builtin(__builtin_amdgcn_wmma_f32_16x16x4_f32)
    v8f acc = {0.f, 0.f, 0.f, 0.f, 0.f, 0.f, 0.f, 0.f};
    for (int k0 = 0; k0 < K; k0 += 4) {
        const int kk = k0 + 2 * half;
        v2f a, b;
        a.x = vec[kk];                    // A replicated across all 16 M rows
        a.y = vec[kk + 1];
        const float* wp = W + (size_t)kk * N + col;
        b.x = wp[0];                      // B[kk,   col]
        b.y = wp[N];                      // B[kk+1, col]
        acc = __builtin_amdgcn_wmma_f32_16x16x4_f32(
            /*neg_a=*/false, a, /*neg_b=*/false, b,
            /*c_mod=*/(short)0, acc, /*reuse_a=*/false, /*reuse_b=*/false);
    }
    float dot = acc[0];                   // D[M=0, N=ln] on lanes 0..15
#else
    float dot = 0.f;
    if (half == 0)
        for (int k = 0; k < K; ++k)
            dot = fmaf(vec[k], W[(size_t)k * N + col], dot);
#endif

    if (half == 0) {
        float v = fmaxf(dot + bias[col], 0.f);
        if (emb) v += emb[(size_t)ident[0] * E_DIM + col];
        out[col] = v;
    }
}

// ---------------------------------------------------------------------------
// gates[j] = x . W_ih[j,:] + h0 . W_hh[j,:] + b_ih[j] + b_hh[j]
// Wave per row j; lanes stream the contiguous row with coalesced float4 loads.
// ---------------------------------------------------------------------------
__global__ void gates_kernel(const float* __restrict__ x, const float* __restrict__ h0,
                             const float* __restrict__ W_ih, const float* __restrict__ W_hh,
                             const float* __restrict__ b_ih, const float* __restrict__ b_hh,
                             float* __restrict__ gates)
{
    const int lane = threadIdx.x & 31;
    const int j = blockIdx.x * (blockDim.x >> 5) + (threadIdx.x >> 5);
    if (j >= G_DIM) return;
    const float4* wi = (const float4*)(W_ih + (size_t)j * E_DIM);
    const float4* wh = (const float4*)(W_hh + (size_t)j * H_DIM);
    const float4* xv = (const float4*)x;
    const float4* hv = (const float4*)h0;
    float acc = 0.f;
#pragma unroll 4
    for (int t = lane; t < E_DIM / 4; t += 32) {
        float4 w = wi[t], v = xv[t];
        acc = fmaf(w.x, v.x, acc); acc = fmaf(w.y, v.y, acc);
        acc = fmaf(w.z, v.z, acc); acc = fmaf(w.w, v.w, acc);
    }
#pragma unroll 4
    for (int t = lane; t < H_DIM / 4; t += 32) {
        float4 w = wh[t], v = hv[t];
        acc = fmaf(w.x, v.x, acc); acc = fmaf(w.y, v.y, acc);
        acc = fmaf(w.z, v.z, acc); acc = fmaf(w.w, v.w, acc);
    }
    for (int off = 16; off; off >>= 1) acc += __shfl_down(acc, off, 32);
    if (lane == 0) gates[j] = acc + b_ih[j] + b_hh[j];
}

// ---------------------------------------------------------------------------
// LSTM cell elementwise (torch gate order i,f,g,o)
// ---------------------------------------------------------------------------
__global__ void lstm_cell_kernel(const float* __restrict__ gates, const float* __restrict__ c0,
                                 float* __restrict__ h_out, float* __restrict__ c_out)
{
    const int t = blockIdx.x * blockDim.x + threadIdx.x;
    if (t >= H_DIM) return;
    const float gi = gates[t];
    const float gf = gates[H_DIM + t];
    const float gg = gates[2 * H_DIM + t];
    const float go = gates[3 * H_DIM + t];
    const float si = 1.f / (1.f + expf(-gi));
    const float sf = 1.f / (1.f + expf(-gf));
    const float so = 1.f / (1.f + expf(-go));
    const float c = sf * c0[t] + si * tanhf(gg);
    const float h = so * tanhf(c);
    h_out[t] = h;
    c_out[t] = c;
}

// ---------------------------------------------------------------------------
// Dominant stream: P[kg][n] = sum_{k in kg-slice} h[k] * W_out[k, n]
// k-split x4 for occupancy on the 400 MB stream; lane-coalesced float4 loads
// (W_out rows are 200000 B -> 64B aligned). h slice staged in LDS.
// ---------------------------------------------------------------------------
__global__ void wout_partial_kernel(const float* __restrict__ h, const float* __restrict__ W,
                                    float* __restrict__ P)
{
    __shared__ float sh[512];
    const int kg = blockIdx.y;
    for (int t = threadIdx.x; t < 512; t += blockDim.x) sh[t] = h[kg * 512 + t];
    __syncthreads();
    const int n4 = blockIdx.x * blockDim.x + threadIdx.x;
    if (n4 >= RULES_N / 4) return;
    const float* col = W + (size_t)kg * 512 * RULES_N + 4 * n4;
    float4 acc = make_float4(0.f, 0.f, 0.f, 0.f);
    for (int k = 0; k < 512; k += 8) {
#pragma unroll
        for (int u = 0; u < 8; ++u) {
            const float hv = sh[k + u];
            const float4 w = *(const float4*)(col + (size_t)(k + u) * RULES_N);
            acc.x = fmaf(hv, w.x, acc.x); acc.y = fmaf(hv, w.y, acc.y);
            acc.z = fmaf(hv, w.z, acc.z); acc.w = fmaf(hv, w.w, acc.w);
        }
    }
    *(float4*)(P + (size_t)kg * RULES_N + 4 * n4) = acc;
}

__global__ void wout_combine_kernel(const float* __restrict__ P, const float* __restrict__ b,
                                    float* __restrict__ z)
{
    const int n = blockIdx.x * blockDim.x + threadIdx.x;
    if (n >= RULES_N) return;
    const float v = P[n] + P[RULES_N + n] + P[2 * RULES_N + n] + P[3 * RULES_N + n] + b[n];
    z[n] = fmaxf(v, 0.f);
}

// ---------------------------------------------------------------------------
// logsumexp over z[0..49999]: red[0]=max, red[1]=log(sum exp(z-max))
// ---------------------------------------------------------------------------
__global__ void logsumexp_kernel(const float* __restrict__ z, float* __restrict__ red)
{
    __shared__ float tmp[32];
    const int tid = threadIdx.x;
    float m = -3.402823466e38f;
    for (int i = tid; i < RULES_N; i += 1024) m = fmaxf(m, z[i]);
    for (int o = 16; o; o >>= 1) m = fmaxf(m, __shfl_down(m, o, 32));
    if ((tid & 31) == 0) tmp[tid >> 5] = m;
    __syncthreads();
    if (tid < 32) {
        float v = tmp[tid];
        for (int o = 16; o; o >>= 1) v = fmaxf(v, __shfl_down(v, o, 32));
        if (tid == 0) tmp[0] = v;
    }
    __syncthreads();
    const float mm = tmp[0];
    __syncthreads();
    float s = 0.f;
    for (int i = tid; i < RULES_N; i += 1024) s += expf(z[i] - mm);
    for (int o = 16; o; o >>= 1) s += __shfl_down(s, o, 32);
    if ((tid & 31) == 0) tmp[tid >> 5] = s;
    __syncthreads();
    if (tid == 0) {
        float t = 0.f;
        for (int i = 0; i < 32; ++i) t += tmp[i];
        red[0] = mm;
        red[1] = logf(t);
    }
}

__global__ void logsoftmax_kernel(const float* __restrict__ z, const float* __restrict__ red,
                                  float* __restrict__ out)
{
    const int n = blockIdx.x * blockDim.x + threadIdx.x;
    if (n >= RULES_N) return;
    out[n] = z[n] - red[0] - red[1];
}

// ---------------------------------------------------------------------------

extern "C" void kernel_launch(void* const* d_in, const int* in_sizes, int n_in,
                              void* d_out, int out_size, void* d_ws, size_t ws_size,
                              hipStream_t stream)
{
    (void)in_sizes; (void)n_in; (void)out_size; (void)ws_size;

    const int*   ident     = (const int*)  d_in[0];
    const float* remainder = (const float*)d_in[1];
    const float* h0        = (const float*)d_in[2];
    const float* c0        = (const float*)d_in[3];
    const float* emb       = (const float*)d_in[4];
    const float* W_rin     = (const float*)d_in[5];
    const float* b_rin     = (const float*)d_in[6];
    const float* W_ih      = (const float*)d_in[7];
    const float* W_hh      = (const float*)d_in[8];
    const float* b_ih      = (const float*)d_in[9];
    const float* b_hh      = (const float*)d_in[10];
    const float* W_ro      = (const float*)d_in[11];
    const float* b_ro      = (const float*)d_in[12];
    const float* W_out     = (const float*)d_in[13];
    const float* b_out     = (const float*)d_in[14];

    float* ws    = (float*)d_ws;
    float* x     = ws + WS_X;
    float* gates = ws + WS_GATES;
    float* z     = ws + WS_Z;
    float* red   = ws + WS_RED;
    float* P     = ws + WS_P;

    float* out_lsm = (float*)d_out;          // 50000
    float* h_out   = out_lsm + RULES_N;      // 2048
    float* c_out   = h_out + H_DIM;          // 2048
    float* rem_out = c_out + H_DIM;          // 1024

    // x = emb[ident] + relu(remainder @ W_rin + b_rin)   (WMMA f32 path)
    gemv_kn_relu_kernel<<<16, 128, 0, stream>>>(W_rin, remainder, b_rin, emb, ident,
                                                x, R_DIM, E_DIM);
    // gates = x @ W_ih.T + h0 @ W_hh.T + b_ih + b_hh     (96 MB stream, 8192 waves)
    gates_kernel<<<G_DIM / 8, 256, 0, stream>>>(x, h0, W_ih, W_hh, b_ih, b_hh, gates);
    // h, c
    lstm_cell_kernel<<<H_DIM / 256, 256, 0, stream>>>(gates, c0, h_out, c_out);
    // remainder_out = relu(h @ W_ro + b_ro)              (WMMA f32 path)
    gemv_kn_relu_kernel<<<16, 128, 0, stream>>>(W_ro, h_out, b_ro, nullptr, nullptr,
                                                rem_out, H_DIM, R_DIM);
    // z = relu(h @ W_out + b_out)                        (400 MB stream, k-split x4)
    dim3 g5((RULES_N / 4 + 255) / 256, 4);
    wout_partial_kernel<<<g5, 256, 0, stream>>>(h_out, W_out, P);
    wout_combine_kernel<<<(RULES_N + 255) / 256, 256, 0, stream>>>(P, b_out, z);
    // out = z - max - log(sum exp(z - max))
    logsumexp_kernel<<<1, 1024, 0, stream>>>(z, red);
    logsoftmax_kernel<<<(RULES_N + 255) / 256, 256, 0, stream>>>(z, red, out_lsm);
}